// MonoSIMoptimizerTorch_31361851196003
// MI455X (gfx1250) — compile-verified
//
#include <hip/hip_runtime.h>

// ---------------------------------------------------------------------------
// MonoSIM optical chain: G = prod_{l=5..0} diag(exp(i*xi_l)) @ W_l  (complex64)
//
// HBM-bound (~604 MB weights @ 23.3 TB/s ~= 26us; only ~0.58 TFLOP of math),
// so we keep full fp32 precision and use V_WMMA_F32_16X16X4_F32.
// Intermediates live in L2 (33.5 MB planes << 192 MB L2).
// ---------------------------------------------------------------------------

typedef float v2f __attribute__((ext_vector_type(2)));
typedef float v8f __attribute__((ext_vector_type(8)));

#define N_IN  1024
#define M_INT 4096
#define N_OUT 1024

#define BM 128   // block tile rows   (4 waves of 32)
#define BN 64    // block tile cols   (2 waves of 32)

__device__ __forceinline__ v8f wmma_f32(v2f a, v2f b, v8f c) {
    // D = A(16x4 f32) * B(4x16 f32) + C(16x16 f32)
    return __builtin_amdgcn_wmma_f32_16x16x4_f32(
        /*neg_a=*/false, a, /*neg_b=*/false, b,
        /*c_mod=*/(short)0, c, /*reuse_a=*/false, /*reuse_b=*/false);
}

// Layer 0: G1 = diag(exp(i*xi0)) * W0   (elementwise row-phase, de-interleave)
__global__ __launch_bounds__(256)
void phase_scale_w0(const float* __restrict__ W0,   // interleaved complex [M_INT x N_IN]
                    const float* __restrict__ xi,   // [M_INT]
                    float* __restrict__ Gre, float* __restrict__ Gim)
{
    int idx = blockIdx.x * blockDim.x + threadIdx.x;
    if (idx >= M_INT * N_IN) return;
    int r = idx >> 10;                     // N_IN == 1024
    float s, c;
    __sincosf(xi[r], &s, &c);
    float re = W0[2 * idx];
    float im = W0[2 * idx + 1];
    Gre[idx] = c * re - s * im;
    Gim[idx] = s * re + c * im;
}

// Epilogue helper: apply row phase and store one 16x16 fragment's 8 rows.
template <bool INTERLEAVED>
__device__ __forceinline__ void emit_tile(float* __restrict__ Ore,
                                          float* __restrict__ Oim,
                                          const float* __restrict__ xi,
                                          int row0, int col, int N,
                                          v8f re8, v8f im8)
{
#pragma unroll
    for (int v = 0; v < 8; ++v) {
        int row = row0 + v;
        float s, c;
        __sincosf(xi[row], &s, &c);
        float re = re8[v], im = im8[v];
        float ore = c * re - s * im;
        float oim = s * re + c * im;
        if (INTERLEAVED) {
            size_t o = ((size_t)row * N + col) * 2;
            Ore[o]     = ore;
            Ore[o + 1] = oim;
        } else {
            size_t o = (size_t)row * N + col;
            Ore[o] = ore;
            Oim[o] = oim;
        }
    }
}

// C = diag(exp(i*xi)) * (W @ B); W interleaved complex [M x K], B as planes
// [K x N], output either planes or interleaved complex64.
template <bool INTERLEAVED>
__global__ __launch_bounds__(256)
void cgemm_rowphase(const float* __restrict__ W,
                    const float* __restrict__ Bre,
                    const float* __restrict__ Bim,
                    const float* __restrict__ xi,
                    float* __restrict__ Ore,
                    float* __restrict__ Oim,
                    int M, int K, int N)
{
    const int lane = threadIdx.x & 31;
    const int wave = threadIdx.x >> 5;
    const int wm   = wave & 3;             // 4 waves down M
    const int wn   = wave >> 2;            // 2 waves across N
    const int rowBase = blockIdx.x * BM + wm * 32;
    const int colBase = blockIdx.y * BN + wn * 32;
    const int l15 = lane & 15;
    const int hi  = lane >> 4;             // 0 for lanes 0-15, 1 for 16-31
    const int kh  = hi * 2;                // K offset for upper half-wave

    // Accumulators: 2x2 fragments x {re, im} = 64 VGPRs / lane
    v8f accRe00 = {0,0,0,0,0,0,0,0}, accRe01 = {0,0,0,0,0,0,0,0};
    v8f accRe10 = {0,0,0,0,0,0,0,0}, accRe11 = {0,0,0,0,0,0,0,0};
    v8f accIm00 = {0,0,0,0,0,0,0,0}, accIm01 = {0,0,0,0,0,0,0,0};
    v8f accIm10 = {0,0,0,0,0,0,0,0}, accIm11 = {0,0,0,0,0,0,0,0};

    // Per-lane A base pointers (ISA A-layout: lanes 0-15 rows M, VGPR = K
    // parity, lanes 16-31 hold K+2/K+3). One float4 = {re(k),im(k),re(k+1),
    // im(k+1)}, 16B aligned because k and kh keep the float offset % 4 == 0.
    const float* Wp0 = W + ((size_t)(rowBase + l15)      * K + kh) * 2;
    const float* Wp1 = W + ((size_t)(rowBase + 16 + l15) * K + kh) * 2;
    const int c0 = colBase + l15;
    const int c1 = colBase + 16 + l15;

#pragma unroll 2
    for (int k = 0; k < K; k += 4) {
        // Prefetch streamed W rows ~1KB ahead (global_prefetch_b8).
        __builtin_prefetch(Wp0 + 2 * k + 256, 0, 0);
        __builtin_prefetch(Wp1 + 2 * k + 256, 0, 0);

        // ---- A fragments (16x4, complex) ----
        const float4 fa0 = *(const float4*)(Wp0 + 2 * k);
        const float4 fa1 = *(const float4*)(Wp1 + 2 * k);
        v2f aRe0 = {fa0.x, fa0.z}, aIm0 = {fa0.y, fa0.w};
        v2f aRe1 = {fa1.x, fa1.z}, aIm1 = {fa1.y, fa1.w};
        v2f aImN0 = -aIm0, aImN1 = -aIm1;   // f32 WMMA has no A-neg modifier

        // ---- B fragments (4x16): VGPR0 = row k+kh, VGPR1 = row k+kh+1 ----
        const int kr = k + kh;
        const float* br = Bre + (size_t)kr * N;
        const float* bi = Bim + (size_t)kr * N;
        v2f bRe0 = {br[c0], br[N + c0]};
        v2f bRe1 = {br[c1], br[N + c1]};
        v2f bIm0 = {bi[c0], bi[N + c0]};
        v2f bIm1 = {bi[c1], bi[N + c1]};

        // ---- complex MACs: 4 WMMAs per output fragment ----
        accRe00 = wmma_f32(aRe0,  bRe0, accRe00);
        accRe00 = wmma_f32(aImN0, bIm0, accRe00);
        accIm00 = wmma_f32(aRe0,  bIm0, accIm00);
        accIm00 = wmma_f32(aIm0,  bRe0, accIm00);

        accRe01 = wmma_f32(aRe0,  bRe1, accRe01);
        accRe01 = wmma_f32(aImN0, bIm1, accRe01);
        accIm01 = wmma_f32(aRe0,  bIm1, accIm01);
        accIm01 = wmma_f32(aIm0,  bRe1, accIm01);

        accRe10 = wmma_f32(aRe1,  bRe0, accRe10);
        accRe10 = wmma_f32(aImN1, bIm0, accRe10);
        accIm10 = wmma_f32(aRe1,  bIm0, accIm10);
        accIm10 = wmma_f32(aIm1,  bRe0, accIm10);

        accRe11 = wmma_f32(aRe1,  bRe1, accRe11);
        accRe11 = wmma_f32(aImN1, bIm1, accRe11);
        accIm11 = wmma_f32(aRe1,  bIm1, accIm11);
        accIm11 = wmma_f32(aIm1,  bRe1, accIm11);
    }

    // C/D layout: lane 0-15 -> row = VGPR v; lane 16-31 -> row = v + 8.
    const int r0 = rowBase + hi * 8;
    emit_tile<INTERLEAVED>(Ore, Oim, xi, r0,      c0, N, accRe00, accIm00);
    emit_tile<INTERLEAVED>(Ore, Oim, xi, r0,      c1, N, accRe01, accIm01);
    emit_tile<INTERLEAVED>(Ore, Oim, xi, r0 + 16, c0, N, accRe10, accIm10);
    emit_tile<INTERLEAVED>(Ore, Oim, xi, r0 + 16, c1, N, accRe11, accIm11);
}

extern "C" void kernel_launch(void* const* d_in, const int* in_sizes, int n_in,
                              void* d_out, int out_size, void* d_ws, size_t ws_size,
                              hipStream_t stream) {
    (void)in_sizes; (void)n_in; (void)out_size; (void)ws_size;

    const float* xi[6];
    const float* W[6];
    for (int l = 0; l < 6; ++l) xi[l] = (const float*)d_in[l];
    for (int l = 0; l < 6; ++l) W[l]  = (const float*)d_in[6 + l];
    float* out = (float*)d_out;   // interleaved complex64 [N_OUT x N_IN]

    // Workspace: two ping-pong de-interleaved plane pairs (4 x 16 MB = 64 MB).
    const size_t PLANE = (size_t)M_INT * N_IN;
    float* A_re = (float*)d_ws;
    float* A_im = A_re + PLANE;
    float* B_re = A_im + PLANE;
    float* B_im = B_re + PLANE;

    // Layer 0: phase * W0 -> A planes
    int total = M_INT * N_IN;
    phase_scale_w0<<<(total + 255) / 256, 256, 0, stream>>>(W[0], xi[0], A_re, A_im);

    dim3 blk(256);
    dim3 gmid(M_INT / BM, N_IN / BN);   // (32, 16)
    cgemm_rowphase<false><<<gmid, blk, 0, stream>>>(W[1], A_re, A_im, xi[1],
                                                    B_re, B_im, M_INT, M_INT, N_IN);
    cgemm_rowphase<false><<<gmid, blk, 0, stream>>>(W[2], B_re, B_im, xi[2],
                                                    A_re, A_im, M_INT, M_INT, N_IN);
    cgemm_rowphase<false><<<gmid, blk, 0, stream>>>(W[3], A_re, A_im, xi[3],
                                                    B_re, B_im, M_INT, M_INT, N_IN);
    cgemm_rowphase<false><<<gmid, blk, 0, stream>>>(W[4], B_re, B_im, xi[4],
                                                    A_re, A_im, M_INT, M_INT, N_IN);

    dim3 gfin(N_OUT / BM, N_IN / BN);   // (8, 16)
    cgemm_rowphase<true><<<gfin, blk, 0, stream>>>(W[5], A_re, A_im, xi[5],
                                                   out, nullptr, N_OUT, M_INT, N_IN);
}